// DAAttention_15522011807793
// MI455X (gfx1250) — compile-verified
//
#include <hip/hip_runtime.h>

// ---------------------------------------------------------------------------
// DAAttention for MI455X (gfx1250, wave32, WMMA).
// All GEMMs use V_WMMA_F32_16X16X4_F32 (fp32, matches reference precision).
// The 288MB e_pam tensor is never materialized: each workgroup keeps a
// 32x1024 f32 score stripe in LDS (CDNA5's 320KB LDS makes this possible),
// softmaxes it in place, and fuses the P@V^T WMMA accumulation (flash-style).
// Global<->LDS staging uses b128 (float4) wherever 16B-aligned; the tgts
// output slab is odd-offset (alpha is 9 floats), so T-sourced reads take a
// runtime-uniform scalar fallback. ~25 GFLOP vs ~100MB HBM -> memory roofline.
// ---------------------------------------------------------------------------

typedef __attribute__((ext_vector_type(2))) float v2f;
typedef __attribute__((ext_vector_type(8))) float v8f;

#define NNUM 1024
#define EPITCH 1028   // 1028%64==4 -> PV A-frag reads hit distinct banks; rows 16B aligned
#define KVPITCH 68    // 68%64==4  -> both K (row) and V (transposed) reads bank-clean; 16B aligned

__device__ __forceinline__ v8f v8zero() {
  v8f z = {0.f, 0.f, 0.f, 0.f, 0.f, 0.f, 0.f, 0.f};
  return z;
}

// acc += A(16 x 64, row-major, lda) * B(64 x 16, row-major, ldb), fp32 WMMA.
// A fragment: lane l holds M=l%16, K = 2*(l/16)+r  (32-bit A 16x4 layout)
// B fragment: lane l holds N=l%16, K = 2*(l/16)+r
__device__ __forceinline__ v8f wmma_k64(const float* A, int lda,
                                        const float* B, int ldb, v8f acc) {
  const int lane = threadIdx.x & 31;
  const int m = lane & 15;
  const int kh = (lane >> 4) << 1;
#pragma unroll
  for (int k0 = 0; k0 < 64; k0 += 4) {
    v2f a, b;
    a.x = A[m * lda + k0 + kh];
    a.y = A[m * lda + k0 + kh + 1];
    b.x = B[(k0 + kh) * ldb + m];
    b.y = B[(k0 + kh + 1) * ldb + m];
    acc = __builtin_amdgcn_wmma_f32_16x16x4_f32(false, a, false, b,
                                                (short)0, acc, false, false);
  }
  return acc;
}

// Same but B is given transposed in LDS: B[k][n] = Bt[n*ldbt + k]
__device__ __forceinline__ v8f wmma_k64_bt(const float* A, int lda,
                                           const float* Bt, int ldbt, v8f acc) {
  const int lane = threadIdx.x & 31;
  const int m = lane & 15;
  const int kh = (lane >> 4) << 1;
#pragma unroll
  for (int k0 = 0; k0 < 64; k0 += 4) {
    v2f a, b;
    a.x = A[m * lda + k0 + kh];
    a.y = A[m * lda + k0 + kh + 1];
    b.x = Bt[m * ldbt + k0 + kh];
    b.y = Bt[m * ldbt + k0 + kh + 1];
    acc = __builtin_amdgcn_wmma_f32_16x16x4_f32(false, a, false, b,
                                                (short)0, acc, false, false);
  }
  return acc;
}

// Stage a 64x64 f32 block (row pitch srcPitch) into LDS at pitch 68.
// b128 path when the source is 16B aligned (srcPitch must be mult of 4).
__device__ __forceinline__ void stage64x64(const float* __restrict__ src,
                                           long srcPitch, float* __restrict__ dst,
                                           int tid) {
  if ((((unsigned long)src) & 15) == 0) {
    for (int idx = tid; idx < 1024; idx += 256) {
      int c = idx >> 4, k = (idx & 15) << 2;
      float4 v = *(const float4*)(src + (long)c * srcPitch + k);
      *(float4*)(dst + c * KVPITCH + k) = v;
    }
  } else {
    for (int idx = tid; idx < 4096; idx += 256) {
      int c = idx >> 6, k = idx & 63;
      dst[c * KVPITCH + k] = src[(long)c * srcPitch + k];
    }
  }
}

// ---------------------------------------------------------------------------
// Generic 64-output-channel GEMM: Y_g[64][1024] = W_g[64 x Cin] @ X_g[Cin x 1024] (+bias)
// Grid: (G, 8).  Block: 256 (8 waves).  Tile per block: 64 x 128.
// X offset  = (g/xdiv)*xhi + (g%xdiv)*xlo  (handles the interleaved tgt layout)
// W index   = g / wdiv
// Covers fs1/fs2/ft1/ft2 (1x1 convs), q/k/v/cv projections and CAM output GEMM.
// ---------------------------------------------------------------------------
__global__ void __launch_bounds__(256) gemm_oc64(
    const float* __restrict__ X, long xhi, long xlo, int xdiv,
    const float* __restrict__ W, long wstride, int wdiv,
    const float* __restrict__ bias, int bstride,
    float* __restrict__ Y, long ystride, int Cin) {
  __shared__ float Ws[64 * KVPITCH];  // A chunk [64 x 64]
  __shared__ float Xs[64 * 132];      // B chunk [64 x 128], 132*4%16==0
  const int g = blockIdx.x;
  const int n0 = blockIdx.y * 128;
  const float* Xg = X + (long)(g / xdiv) * xhi + (long)(g % xdiv) * xlo;
  const int widx = g / wdiv;
  const float* Wg = W + (long)widx * wstride;
  float* Yg = Y + (long)g * ystride;
  const int tid = threadIdx.x;
  const int wave = tid >> 5, lane = tid & 31;
  const bool xAligned = ((((unsigned long)Xg) & 15) == 0);

  v8f acc[4];
#pragma unroll
  for (int mt = 0; mt < 4; ++mt) acc[mt] = v8zero();

  for (int c0 = 0; c0 < Cin; c0 += 64) {
    stage64x64(Wg + c0, Cin, Ws, tid);  // weights always 16B aligned
    const float* Xc = Xg + (long)c0 * NNUM + n0;
    if (xAligned) {
      for (int idx = tid; idx < 2048; idx += 256) {
        int kk = idx >> 5, n = (idx & 31) << 2;
        float4 v = *(const float4*)(Xc + (long)kk * NNUM + n);
        *(float4*)(&Xs[kk * 132 + n]) = v;
      }
    } else {
      for (int idx = tid; idx < 8192; idx += 256) {
        int kk = idx >> 7, n = idx & 127;
        Xs[kk * 132 + n] = Xc[(long)kk * NNUM + n];
      }
    }
    __syncthreads();
#pragma unroll
    for (int mt = 0; mt < 4; ++mt)
      acc[mt] = wmma_k64(&Ws[mt * 16 * KVPITCH], KVPITCH, &Xs[wave * 16], 132, acc[mt]);
    __syncthreads();
  }
#pragma unroll
  for (int mt = 0; mt < 4; ++mt) {
#pragma unroll
    for (int v = 0; v < 8; ++v) {
      int row = mt * 16 + (lane >> 4) * 8 + v;
      int col = n0 + wave * 16 + (lane & 15);
      float val = acc[mt][v];
      if (bias) val += bias[widx * bstride + row];
      Yg[(long)row * NNUM + col] = val;
    }
  }
}

// ---------------------------------------------------------------------------
// 3x3 stride-2 pad-1 conv (128 -> 64 ch, 64x64 -> 32x32) as implicit GEMM:
// 9 taps x 2 K-chunks of 64, input tile gathered into LDS, WMMA accumulate.
// Grid: (B, 8). Block: 256.  (Gather is conditional per element -> scalar.)
// ---------------------------------------------------------------------------
__global__ void __launch_bounds__(256) conv3x3s2(
    const float* __restrict__ X, const float* __restrict__ W,
    const float* __restrict__ bias, float* __restrict__ Y, long ystride) {
  __shared__ float Ws[64 * KVPITCH];
  __shared__ float Xs[64 * 132];
  const int b = blockIdx.x;
  const int n0 = blockIdx.y * 128;
  const int tid = threadIdx.x, wave = tid >> 5, lane = tid & 31;

  v8f acc[4];
#pragma unroll
  for (int mt = 0; mt < 4; ++mt) acc[mt] = v8zero();

  for (int tap = 0; tap < 9; ++tap) {
    const int dy = tap / 3 - 1, dx = tap % 3 - 1;
    for (int c0 = 0; c0 < 128; c0 += 64) {
      for (int idx = tid; idx < 64 * 64; idx += 256) {
        int o = idx >> 6, kk = idx & 63;
        Ws[o * KVPITCH + kk] = W[(long)(o * 128 + c0 + kk) * 9 + tap];
      }
      for (int idx = tid; idx < 64 * 128; idx += 256) {
        int kk = idx >> 7, p = idx & 127;
        int pp = n0 + p;
        int y = pp >> 5, x = pp & 31;
        int iy = 2 * y + dy, ix = 2 * x + dx;
        float val = 0.f;
        if (iy >= 0 && iy < 64 && ix >= 0 && ix < 64)
          val = X[(((long)b * 128 + c0 + kk) * 64 + iy) * 64 + ix];
        Xs[kk * 132 + p] = val;
      }
      __syncthreads();
#pragma unroll
      for (int mt = 0; mt < 4; ++mt)
        acc[mt] = wmma_k64(&Ws[mt * 16 * KVPITCH], KVPITCH, &Xs[wave * 16], 132, acc[mt]);
      __syncthreads();
    }
  }
  float* Yg = Y + (long)b * ystride;
#pragma unroll
  for (int mt = 0; mt < 4; ++mt) {
#pragma unroll
    for (int v = 0; v < 8; ++v) {
      int row = mt * 16 + (lane >> 4) * 8 + v;
      int col = n0 + wave * 16 + (lane & 15);
      Yg[(long)row * NNUM + col] = acc[mt][v] + bias[row];
    }
  }
}

// ---------------------------------------------------------------------------
// CAM energies: E[ij,b][c][d] = sum_n T[i,b,c,n] * S[j,b,d,n].  Grid: 72 blocks.
// ---------------------------------------------------------------------------
__global__ void __launch_bounds__(256) cam_e_kernel(
    const float* __restrict__ Sbase, const float* __restrict__ Tbase,
    long t_hi, long t_lo, float* __restrict__ Ecam) {
  __shared__ float Ts[64 * KVPITCH];
  __shared__ float Ss[64 * KVPITCH];
  const int idx0 = blockIdx.x;
  const int b = idx0 & 7;
  const int ij = idx0 >> 3;
  const int i = ij / 3, j = ij % 3;
  const float* Tp = Tbase + (long)i * t_hi + (long)b * t_lo;
  const float* Sp = Sbase + (long)(j * 8 + b) * 65536;
  const int tid = threadIdx.x, wave = tid >> 5, lane = tid & 31;

  v8f acc[2];
  acc[0] = v8zero();
  acc[1] = v8zero();

  for (int n0 = 0; n0 < NNUM; n0 += 64) {
    stage64x64(Tp + n0, NNUM, Ts, tid);
    stage64x64(Sp + n0, NNUM, Ss, tid);
    __syncthreads();
#pragma unroll
    for (int r = 0; r < 2; ++r) {
      int t = wave * 2 + r;
      int mt = t >> 2, nt = t & 3;
      acc[r] = wmma_k64_bt(&Ts[mt * 16 * KVPITCH], KVPITCH,
                           &Ss[nt * 16 * KVPITCH], KVPITCH, acc[r]);
    }
    __syncthreads();
  }
#pragma unroll
  for (int r = 0; r < 2; ++r) {
    int t = wave * 2 + r;
    int mt = t >> 2, nt = t & 3;
#pragma unroll
    for (int v = 0; v < 8; ++v) {
      int row = mt * 16 + (lane >> 4) * 8 + v;
      int col = nt * 16 + (lane & 15);
      Ecam[(long)idx0 * 4096 + row * 64 + col] = acc[r][v];
    }
  }
}

// attn_cam[i,b] = sum_j softmax_rows(e_cam[i,j,b]).  Grid 24, block 64.
__global__ void __launch_bounds__(64) cam_attn_kernel(
    const float* __restrict__ Ecam, float* __restrict__ Attn) {
  const int g = blockIdx.x;
  const int i = g >> 3, b = g & 7;
  const int r = threadIdx.x;  // row c
  float acc[64];
#pragma unroll
  for (int d = 0; d < 64; ++d) acc[d] = 0.f;
  for (int j = 0; j < 3; ++j) {
    const float* row = Ecam + (long)((i * 3 + j) * 8 + b) * 4096 + r * 64;
    float mx = -3.402823466e38f;
#pragma unroll
    for (int d = 0; d < 64; ++d) mx = fmaxf(mx, row[d]);
    float s = 0.f;
#pragma unroll
    for (int d = 0; d < 64; ++d) s += __expf(row[d] - mx);
    float inv = 1.f / s;
#pragma unroll
    for (int d = 0; d < 64; ++d) acc[d] += __expf(row[d] - mx) * inv;
  }
  float* out = Attn + (long)g * 4096 + r * 64;
#pragma unroll
  for (int d = 0; d < 64; ++d) out[d] = acc[d];
}

// ---------------------------------------------------------------------------
// PAM flash kernel. Grid: (24 = L*B, 32 = N/32). Block 256 (8 waves).
// Per block: rows [n0, n0+32) of e = Q^T K_j, full 1024 columns in LDS,
// softmax in place (float4 sweeps), accumulate out += P_j @ V^T over j.
// Dynamic LDS ~159 KB -> 2 workgroups per WGP (320 KB pool).
// ---------------------------------------------------------------------------
__global__ void __launch_bounds__(256) pam_kernel(
    const float* __restrict__ Q, const float* __restrict__ K,
    const float* __restrict__ V, float* __restrict__ OutPam,
    float* __restrict__ PamPart) {
  extern __shared__ float sm[];
  float* Es = sm;                      // 32 x 1028
  float* Qs = Es + 32 * EPITCH;        // 32 x 68 (Q^T tile)
  float* KVs = Qs + 32 * 68;           // 64 x 68 (K or V chunk)
  float* red = KVs + 64 * KVPITCH;     // 256
  float* rows = red + 256;             // 64: rowmax [0..31], 1/rowsum [32..63]

  const int g = blockIdx.x;
  const int i = g >> 3, b = g & 7;
  const int n0 = blockIdx.y * 32;
  const int tid = threadIdx.x, wave = tid >> 5, lane = tid & 31;
  const int nt = wave >> 2;  // 0..1  (16-row subtile)
  const int qt = wave & 3;   // 0..3  (m-subtile in E phase / c-tile in PV)

  // stage Q^T: Qs[n][c] = q[g][c][n0+n]
  for (int idx = tid; idx < 2048; idx += 256) {
    int c = idx >> 5, n = idx & 31;
    Qs[n * 68 + c] = Q[(long)g * 65536 + (long)c * NNUM + n0 + n];
  }

  v8f acc = v8zero();
  const float* Vg = V + (long)g * 65536;

  for (int j = 0; j < 3; ++j) {
    const float* Kj = K + (long)(j * 8 + b) * 65536;
    // ---- E phase: Es[32][1024] = Q^T K_j ----
    for (int m0 = 0; m0 < NNUM; m0 += 64) {
      __syncthreads();
      stage64x64(Kj + m0, NNUM, KVs, tid);   // always aligned (ws)
      __syncthreads();
      v8f e = v8zero();
      e = wmma_k64(&Qs[nt * 16 * 68], 68, &KVs[qt * 16], KVPITCH, e);
#pragma unroll
      for (int v = 0; v < 8; ++v)
        Es[(nt * 16 + (lane >> 4) * 8 + v) * EPITCH + m0 + qt * 16 + (lane & 15)] = e[v];
    }
    __syncthreads();

    // ---- softmax over 1024 cols per row; 8 threads/row x 128 cols (b128) ----
    const int r = tid >> 3, seg = tid & 7;
    float* Er = Es + r * EPITCH + seg * 128;   // 16B aligned (EPITCH*4, 512)
    float mx = -3.402823466e38f, rawsum = 0.f;
    for (int t2 = 0; t2 < 128; t2 += 4) {
      float4 x = *(const float4*)&Er[t2];
      mx = fmaxf(mx, fmaxf(fmaxf(x.x, x.y), fmaxf(x.z, x.w)));
      rawsum += (x.x + x.y) + (x.z + x.w);
    }
    red[tid] = mx;
    __syncthreads();
    if (tid < 32) {
      float m2 = red[tid * 8];
      for (int s = 1; s < 8; ++s) m2 = fmaxf(m2, red[tid * 8 + s]);
      rows[tid] = m2;
    }
    __syncthreads();
    red[tid] = rawsum;
    __syncthreads();
    if (tid == 0) {  // deterministic partial sum for alpha_pam (no atomics)
      float s = 0.f;
      for (int t2 = 0; t2 < 256; ++t2) s += red[t2];
      PamPart[(i * 3 + j) * 256 + b * 32 + blockIdx.y] = s;
    }
    float rm = rows[r];
    float es = 0.f;
    for (int t2 = 0; t2 < 128; t2 += 4) {
      float4 x = *(const float4*)&Er[t2];
      x.x = __expf(x.x - rm);
      x.y = __expf(x.y - rm);
      x.z = __expf(x.z - rm);
      x.w = __expf(x.w - rm);
      *(float4*)&Er[t2] = x;
      es += (x.x + x.y) + (x.z + x.w);
    }
    __syncthreads();
    red[tid] = es;
    __syncthreads();
    if (tid < 32) {
      float s2 = 0.f;
      for (int s = 0; s < 8; ++s) s2 += red[tid * 8 + s];
      rows[32 + tid] = 1.f / s2;
    }
    __syncthreads();
    float inv = rows[32 + r];
    for (int t2 = 0; t2 < 128; t2 += 4) {
      float4 x = *(const float4*)&Er[t2];
      x.x *= inv; x.y *= inv; x.z *= inv; x.w *= inv;
      *(float4*)&Er[t2] = x;
    }
    __syncthreads();

    // ---- PV phase: acc[32n x 64c] += P[32 x 1024] @ V^T[1024 x 64] ----
    for (int m0 = 0; m0 < NNUM; m0 += 64) {
      stage64x64(Vg + m0, NNUM, KVs, tid);
      __syncthreads();
      acc = wmma_k64_bt(&Es[nt * 16 * EPITCH + m0], EPITCH,
                        &KVs[qt * 16 * KVPITCH], KVPITCH, acc);
      __syncthreads();
    }
  }
  // write out_pam[g][c][n]
#pragma unroll
  for (int v = 0; v < 8; ++v) {
    int nrow = n0 + nt * 16 + (lane >> 4) * 8 + v;
    int ccol = qt * 16 + (lane & 15);
    OutPam[(long)g * 65536 + (long)ccol * NNUM + nrow] = acc[v];
  }
}

// fused = 0.5*(out_pam + out_cam) -> tgts even slots (odd-offset dst -> scalar)
__global__ void __launch_bounds__(256) fuse_kernel(
    const float* __restrict__ P, const float* __restrict__ C_,
    float* __restrict__ Tg) {
  long idx = (long)blockIdx.x * 256 + threadIdx.x;  // 1,572,864 total
  int g = (int)(idx >> 16);
  int within = (int)(idx & 65535);
  int i = g >> 3, b = g & 7;
  long dst = ((long)(2 * i) * 8 + b) * 65536 + within;
  Tg[dst] = 0.5f * (P[idx] + C_[idx]);
}

// alpha = 0.5*(softmax_j(mean e_cam) + softmax_j(mean e_pam))
__global__ void __launch_bounds__(256) alpha_kernel(
    const float* __restrict__ Ecam, const float* __restrict__ PamPart,
    float* __restrict__ AlphaOut) {
  __shared__ float red[256];
  __shared__ float cm[9];
  const int tid = threadIdx.x;
  for (int ij = 0; ij < 9; ++ij) {
    float s = 0.f;
    for (int idx = tid; idx < 32768; idx += 256) s += Ecam[(long)ij * 32768 + idx];
    red[tid] = s;
    __syncthreads();
    if (tid == 0) {
      float t = 0.f;
      for (int u = 0; u < 256; ++u) t += red[u];
      cm[ij] = t / 32768.f;
    }
    __syncthreads();
  }
  if (tid == 0) {
    for (int i = 0; i < 3; ++i) {
      float pmv[3];
      for (int j = 0; j < 3; ++j) {
        float s = 0.f;
        for (int u = 0; u < 256; ++u) s += PamPart[(i * 3 + j) * 256 + u];
        pmv[j] = s * (1.f / 8388608.f);  // / (B*N*N)
      }
      float mxc = fmaxf(fmaxf(cm[i * 3], cm[i * 3 + 1]), cm[i * 3 + 2]);
      float mxp = fmaxf(fmaxf(pmv[0], pmv[1]), pmv[2]);
      float ec[3], ep[3];
      float sc = 0.f, sp = 0.f;
      for (int j = 0; j < 3; ++j) {
        ec[j] = __expf(cm[i * 3 + j] - mxc);
        sc += ec[j];
        ep[j] = __expf(pmv[j] - mxp);
        sp += ep[j];
      }
      for (int j = 0; j < 3; ++j)
        AlphaOut[i * 3 + j] = 0.5f * (ec[j] / sc + ep[j] / sp);
    }
  }
}

// ---------------------------------------------------------------------------
extern "C" void kernel_launch(void* const* d_in, const int* in_sizes, int n_in,
                              void* d_out, int out_size, void* d_ws, size_t ws_size,
                              hipStream_t stream) {
  (void)in_sizes; (void)n_in; (void)out_size; (void)ws_size;
  const float* src0 = (const float*)d_in[0];
  const float* src1 = (const float*)d_in[1];
  const float* src2 = (const float*)d_in[2];
  const float* tgt0 = (const float*)d_in[3];
  const float* tgt1 = (const float*)d_in[4];
  const float* tgt2 = (const float*)d_in[5];
  const float* fs0_w = (const float*)d_in[6];  const float* fs0_b = (const float*)d_in[7];
  const float* fs1_w = (const float*)d_in[8];  const float* fs1_b = (const float*)d_in[9];
  const float* fs2_w = (const float*)d_in[10]; const float* fs2_b = (const float*)d_in[11];
  const float* ft0_w = (const float*)d_in[12]; const float* ft0_b = (const float*)d_in[13];
  const float* ft1_w = (const float*)d_in[14]; const float* ft1_b = (const float*)d_in[15];
  const float* ft2_w = (const float*)d_in[16]; const float* ft2_b = (const float*)d_in[17];
  const float* pq_w = (const float*)d_in[18]; const float* pq_b = (const float*)d_in[19];
  const float* pk_w = (const float*)d_in[20]; const float* pk_b = (const float*)d_in[21];
  const float* pv_w = (const float*)d_in[22]; const float* pv_b = (const float*)d_in[23];
  const float* cv_w = (const float*)d_in[24]; const float* cv_b = (const float*)d_in[25];

  float* out = (float*)d_out;
  float* S = out;                       // srcs = S  [3][8][65536]
  float* alpha = out + 1572864;         // [9]
  float* tg = out + 1572873;            // tgts [6][8][65536]; odd slots = T

  float* ws = (float*)d_ws;
  float* qws   = ws;
  float* kws   = ws + 1572864;
  float* vws   = ws + 3145728;
  float* vcws  = ws + 4718592;
  float* outp  = ws + 6291456;          // out_pam [24][64][1024]
  float* outc  = ws + 7864320;          // out_cam
  float* ecam  = ws + 9437184;          // [9][8][64][64]
  float* attnc = ws + 9732096;          // [24][64][64]
  float* ppart = ws + 9830400;          // [9][256] pam e partial sums

  const long SLOT = 65536;
  float* Tbase = tg + 8 * SLOT;         // layer stride = 16*SLOT, batch stride = SLOT

  // 1. Projection convs -> S (srcs slab) and T (tgts odd slots)
  conv3x3s2<<<dim3(8, 8), 256, 0, stream>>>(src0, fs0_w, fs0_b, S, SLOT);
  conv3x3s2<<<dim3(8, 8), 256, 0, stream>>>(tgt0, ft0_w, ft0_b, Tbase, SLOT);
  gemm_oc64<<<dim3(8, 8), 256, 0, stream>>>(src1, 262144L, 0L, 1, fs1_w, 0L, 8,
                                            fs1_b, 0, S + 8 * SLOT, SLOT, 256);
  gemm_oc64<<<dim3(8, 8), 256, 0, stream>>>(src2, 524288L, 0L, 1, fs2_w, 0L, 8,
                                            fs2_b, 0, S + 16 * SLOT, SLOT, 512);
  gemm_oc64<<<dim3(8, 8), 256, 0, stream>>>(tgt1, 262144L, 0L, 1, ft1_w, 0L, 8,
                                            ft1_b, 0, tg + 24 * SLOT, SLOT, 256);
  gemm_oc64<<<dim3(8, 8), 256, 0, stream>>>(tgt2, 524288L, 0L, 1, ft2_w, 0L, 8,
                                            ft2_b, 0, tg + 40 * SLOT, SLOT, 512);

  // 2. q/k/v/vc 1x1 projections (q,v,vc from T; k from S)
  gemm_oc64<<<dim3(24, 8), 256, 0, stream>>>(Tbase, 16 * SLOT, SLOT, 8, pq_w,
                                             4096L, 8, pq_b, 64, qws, SLOT, 64);
  gemm_oc64<<<dim3(24, 8), 256, 0, stream>>>(S, SLOT, 0L, 1, pk_w,
                                             4096L, 8, pk_b, 64, kws, SLOT, 64);
  gemm_oc64<<<dim3(24, 8), 256, 0, stream>>>(Tbase, 16 * SLOT, SLOT, 8, pv_w,
                                             4096L, 8, pv_b, 64, vws, SLOT, 64);
  gemm_oc64<<<dim3(24, 8), 256, 0, stream>>>(Tbase, 16 * SLOT, SLOT, 8, cv_w,
                                             4096L, 8, cv_b, 64, vcws, SLOT, 64);

  // 3. CAM
  cam_e_kernel<<<dim3(72), 256, 0, stream>>>(S, Tbase, 16 * SLOT, SLOT, ecam);
  cam_attn_kernel<<<dim3(24), 64, 0, stream>>>(ecam, attnc);
  gemm_oc64<<<dim3(24, 8), 256, 0, stream>>>(vcws, SLOT, 0L, 1, attnc, 4096L, 1,
                                             nullptr, 0, outc, SLOT, 64);

  // 4. PAM flash (dynamic LDS ~159 KB -> 2 WGs/WGP)
  const int pamLds = (32 * EPITCH + 32 * 68 + 64 * KVPITCH + 256 + 64) * (int)sizeof(float);
  (void)hipFuncSetAttribute((const void*)pam_kernel,
                            hipFuncAttributeMaxDynamicSharedMemorySize, pamLds);
  pam_kernel<<<dim3(24, 32), 256, pamLds, stream>>>(qws, kws, vws, outp, ppart);

  // 5. Finalize
  alpha_kernel<<<dim3(1), 256, 0, stream>>>(ecam, ppart, alpha);
  fuse_kernel<<<dim3(6144), 256, 0, stream>>>(outp, outc, tg);
}